// GNNEdgeClassifier_75840532513189
// MI455X (gfx1250) — compile-verified
//
#include <hip/hip_runtime.h>

typedef __attribute__((ext_vector_type(2))) float v2f;
typedef __attribute__((ext_vector_type(8))) float v8f;

#define HID 128

// ---------------------------------------------------------------- copy x
__global__ __launch_bounds__(256) void copy_x_kernel(const float* __restrict__ src,
                                                     float* __restrict__ dst, int n4) {
  int i = blockIdx.x * 256 + threadIdx.x;
  if (i < n4) ((float4*)dst)[i] = ((const float4*)src)[i];
}

// ------------------------------------------------ message + atomic scatter
// msg = relu(x[src] + ea*We + be); aggr[dst] += msg   (one wave-lane = 4 feats)
__global__ __launch_bounds__(256) void msg_scatter_kernel(
    const float* __restrict__ x, const float* __restrict__ ea,
    const long long* __restrict__ src, const long long* __restrict__ dst,
    const float* __restrict__ We, const float* __restrict__ be,
    float* __restrict__ aggr, int E_) {
  long long gid = (long long)blockIdx.x * 256 + threadIdx.x;
  long long e = gid >> 5;
  if (e >= E_) return;
  int c = ((int)gid & 31) << 2;
  long long s = src[e], d = dst[e];
  float a = ea[e];
  float4 xv = *(const float4*)(x + (size_t)s * HID + c);
  float4 wv = *(const float4*)(We + c);
  float4 bv = *(const float4*)(be + c);
  float* ap = aggr + (size_t)d * HID + c;
  float m;
  m = xv.x + a * wv.x + bv.x; atomicAdd(ap + 0, m > 0.f ? m : 0.f);
  m = xv.y + a * wv.y + bv.y; atomicAdd(ap + 1, m > 0.f ? m : 0.f);
  m = xv.z + a * wv.z + bv.z; atomicAdd(ap + 2, m > 0.f ? m : 0.f);
  m = xv.w + a * wv.w + bv.w; atomicAdd(ap + 3, m > 0.f ? m : 0.f);
}

// ------------------------------------------------ node GEMM: relu((Xa[+Xb])@W + bias)
// WG = 256 thr = 8 waves, covers 128 rows. Each wave: one 16-row stripe x 128 cols
// via 8 accumulator tiles of V_WMMA_F32_16X16X4_F32. X block and W^T staged in LDS
// (pitch 132 floats -> conflict-free ds_load_b64 fragments).
__global__ __launch_bounds__(256) void gemm_node_kernel(
    const float* __restrict__ Xa, const float* __restrict__ Xb,
    const float* __restrict__ W, const float* __restrict__ bias,
    float* __restrict__ Out, int nrows) {
  extern __shared__ float smem[];
  float* sX  = smem;              // [128][132]
  float* sWt = smem + 128 * 132;  // [128 cols][132 k]  (W transposed)
  const int tid = threadIdx.x;
  const int rowBlock = blockIdx.x * 128;

  for (int i = tid; i < 128 * 32; i += 256) {   // stage X (+aggr) as float4
    int r = i >> 5, c4 = (i & 31) << 2;
    int row = rowBlock + r;
    float4 v = make_float4(0.f, 0.f, 0.f, 0.f);
    if (row < nrows) {
      v = *(const float4*)(Xa + (size_t)row * HID + c4);
      if (Xb) {
        float4 u = *(const float4*)(Xb + (size_t)row * HID + c4);
        v.x += u.x; v.y += u.y; v.z += u.z; v.w += u.w;
      }
    }
    *(float4*)&sX[r * 132 + c4] = v;
  }
  for (int i = tid; i < HID * HID; i += 256) {  // stage W transposed
    int k = i >> 7, c = i & 127;
    sWt[c * 132 + k] = W[i];
  }
  __syncthreads();

  const int lane = tid & 31, wave = tid >> 5;
  const int l16 = lane & 15;
  const int half2 = ((lane >> 4) << 1);         // 0 or 2 (K offset of half-wave)
  const int r0 = wave * 16;

  v8f zero = {0.f, 0.f, 0.f, 0.f, 0.f, 0.f, 0.f, 0.f};
  v8f acc[8];
#pragma unroll
  for (int n = 0; n < 8; ++n) acc[n] = zero;

  for (int k0 = 0; k0 < HID; k0 += 4) {
    v2f a = *(const v2f*)&sX[(r0 + l16) * 132 + k0 + half2];
#pragma unroll
    for (int n = 0; n < 8; ++n) {
      v2f b = *(const v2f*)&sWt[(n * 16 + l16) * 132 + k0 + half2];
      acc[n] = __builtin_amdgcn_wmma_f32_16x16x4_f32(false, a, false, b,
                                                     (short)0, acc[n], false, false);
    }
  }

#pragma unroll
  for (int n = 0; n < 8; ++n) {
    int col = n * 16 + l16;
    float bv = bias[col];
#pragma unroll
    for (int r = 0; r < 8; ++r) {
      int row = rowBlock + r0 + r + (half2 << 2);   // M = r + 8*half
      if (row < nrows) {
        float v = acc[n][r] + bv;
        Out[(size_t)row * HID + col] = v > 0.f ? v : 0.f;
      }
    }
  }
}

// ------------------------------------------------ fused edge predictor
// one wave = 16 edges. A = [x[src] || x[dst]] (16x256) gathered from global,
// B = Wp1^T in LDS (pitch 260). After 8 WMMA tiles: bias+relu, dot with Wp2,
// 16-lane shfl_xor reduction -> scalar per edge. Hidden tensor never hits HBM.
__global__ __launch_bounds__(256) void edge_pred_kernel(
    const float* __restrict__ x,
    const long long* __restrict__ src, const long long* __restrict__ dst,
    const float* __restrict__ Wp1, const float* __restrict__ bp1,
    const float* __restrict__ Wp2, const float* __restrict__ bp2,
    float* __restrict__ out, int E_) {
  extern __shared__ float smem[];
  float* sWt = smem;  // [128 cols][260 k] of Wp1^T
  const int tid = threadIdx.x;
  for (int i = tid; i < 256 * HID; i += 256) {
    int k = i >> 7, c = i & 127;
    sWt[c * 260 + k] = Wp1[i];
  }
  __syncthreads();

  const int lane = tid & 31, wave = tid >> 5;
  const int l16 = lane & 15, half = lane >> 4;
  const int half2 = half << 1;
  const long long e0 = (long long)blockIdx.x * 128 + wave * 16;
  long long eMine = e0 + l16;
  long long eC = eMine < E_ ? eMine : (long long)(E_ - 1);  // clamp: safe loads, guarded stores
  const float* ps = x + (size_t)src[eC] * HID;
  const float* pd = x + (size_t)dst[eC] * HID;

  v8f zero = {0.f, 0.f, 0.f, 0.f, 0.f, 0.f, 0.f, 0.f};
  v8f acc[8];
#pragma unroll
  for (int n = 0; n < 8; ++n) acc[n] = zero;

  for (int k0 = 0; k0 < 2 * HID; k0 += 4) {
    int k = k0 + half2;
    const float* base = (k < HID) ? (ps + k) : (pd + (k - HID));  // concat(x_src, x_dst)
    v2f a; a.x = base[0]; a.y = base[1];
#pragma unroll
    for (int n = 0; n < 8; ++n) {
      v2f b = *(const v2f*)&sWt[(n * 16 + l16) * 260 + k0 + half2];
      acc[n] = __builtin_amdgcn_wmma_f32_16x16x4_f32(false, a, false, b,
                                                     (short)0, acc[n], false, false);
    }
  }

  float bp2v = bp2[0];
  float partial[8];
#pragma unroll
  for (int r = 0; r < 8; ++r) partial[r] = 0.f;
#pragma unroll
  for (int n = 0; n < 8; ++n) {
    int col = n * 16 + l16;
    float b1 = bp1[col], w2 = Wp2[col];
#pragma unroll
    for (int r = 0; r < 8; ++r) {
      float h = acc[n][r] + b1;
      partial[r] += (h > 0.f ? h : 0.f) * w2;
    }
  }
#pragma unroll
  for (int r = 0; r < 8; ++r) {
    partial[r] += __shfl_xor(partial[r], 8, 16);
    partial[r] += __shfl_xor(partial[r], 4, 16);
    partial[r] += __shfl_xor(partial[r], 2, 16);
    partial[r] += __shfl_xor(partial[r], 1, 16);
  }
  if (l16 == 0) {
#pragma unroll
    for (int r = 0; r < 8; ++r) {
      long long eo = e0 + half * 8 + r;   // C-tile M = r + 8*half
      if (eo < E_) out[eo] = partial[r] + bp2v;
    }
  }
}

// ---------------------------------------------------------------- launcher
extern "C" void kernel_launch(void* const* d_in, const int* in_sizes, int n_in,
                              void* d_out, int out_size, void* d_ws, size_t ws_size,
                              hipStream_t stream) {
  (void)n_in; (void)out_size; (void)ws_size;
  const float* x        = (const float*)d_in[0];
  const float* ea       = (const float*)d_in[1];
  const long long* ei   = (const long long*)d_in[2];
  const float* Wl1      = (const float*)d_in[3];
  const float* bl1      = (const float*)d_in[4];
  const float* Wl2      = (const float*)d_in[5];
  const float* bl2      = (const float*)d_in[6];
  const float* We       = (const float*)d_in[7];
  const float* be       = (const float*)d_in[8];
  const float* Wp1      = (const float*)d_in[9];
  const float* bp1      = (const float*)d_in[10];
  const float* Wp2      = (const float*)d_in[11];
  const float* bp2      = (const float*)d_in[12];

  const int N = in_sizes[0] / HID;   // 50000
  const int E = in_sizes[1];         // 800000
  const int L = in_sizes[4] / HID;   // 3

  float* bufA = (float*)d_ws;
  float* bufB = bufA + (size_t)N * HID;
  float* bufC = bufB + (size_t)N * HID;

  {
    int n4 = N * HID / 4;
    copy_x_kernel<<<(n4 + 255) / 256, 256, 0, stream>>>(x, bufA, n4);
  }

  float* xcur = bufA;
  float* scratch = bufB;
  const size_t ldsGemm = (size_t)(128 * 132 + 128 * 132) * sizeof(float);

  for (int l = 0; l < L; ++l) {
    hipMemsetAsync(scratch, 0, (size_t)N * HID * sizeof(float), stream);
    long long mthreads = (long long)E * 32;
    msg_scatter_kernel<<<(int)((mthreads + 255) / 256), 256, 0, stream>>>(
        xcur, ea, ei, ei + E, We + (size_t)l * HID, be + (size_t)l * HID, scratch, E);
    gemm_node_kernel<<<(N + 127) / 128, 256, ldsGemm, stream>>>(
        xcur, scratch, Wl1 + (size_t)l * HID * HID, bl1 + (size_t)l * HID, bufC, N);
    gemm_node_kernel<<<(N + 127) / 128, 256, ldsGemm, stream>>>(
        bufC, nullptr, Wl2 + (size_t)l * HID * HID, bl2 + (size_t)l * HID, scratch, N);
    float* t = xcur; xcur = scratch; scratch = t;
  }

  const size_t ldsPred = (size_t)128 * 260 * sizeof(float);
  edge_pred_kernel<<<(E + 127) / 128, 256, ldsPred, stream>>>(
      xcur, ei, ei + E, Wp1, bp1, Wp2, bp2, (float*)d_out, E);
}